// LIFCell_11364483465332
// MI455X (gfx1250) — compile-verified
//
#include <hip/hip_runtime.h>

typedef __attribute__((ext_vector_type(2))) float v2f;
typedef __attribute__((ext_vector_type(8))) float v8f;

#define UNITS   256
#define N_IN    128
#define N_EXC   51              // int(256 * 0.2)
#define N_REFR  5
#define DECAY_F 0.9512294245007140f   // exp(-1/20)

// ---------------------------------------------------------------------------
// Prep: apply connectivity / EI / no-self-connection constraints to the
// recurrent weights and write BOTH weight matrices directly in WMMA
// B-fragment packed order:
//   fragment for (k4, tile): 32 lanes x float2, lane = (koff>=2)*16 + (n&15),
//   comp = koff&1, where k = 4*k4+koff, n = 16*tile + (n&15).
// ---------------------------------------------------------------------------
__global__ void lif_prep_pack(const float* __restrict__ iw,
                              const float* __restrict__ rw,
                              const unsigned char* __restrict__ conn,
                              float* __restrict__ iwpk,   // 128x256 packed
                              float* __restrict__ wpk)    // 256x256 packed
{
    int id = blockIdx.x * blockDim.x + threadIdx.x;
    if (id < UNITS * UNITS) {
        int k = id >> 8, n = id & 255;
        float val = conn[id] ? rw[id] : 0.0f;
        val = (n < N_EXC) ? fmaxf(val, 0.0f) : fminf(val, 0.0f);
        if (k == n) val = 0.0f;
        int k4 = k >> 2, koff = k & 3, tile = n >> 4, nl = n & 15;
        int lane = ((koff >> 1) << 4) | nl;
        int comp = koff & 1;
        wpk[((((k4 << 4) + tile) << 5) + lane) * 2 + comp] = val;
    } else if (id < UNITS * UNITS + N_IN * UNITS) {
        int id2 = id - UNITS * UNITS;
        int k = id2 >> 8, n = id2 & 255;
        float val = iw[id2];
        int k4 = k >> 2, koff = k & 3, tile = n >> 4, nl = n & 15;
        int lane = ((koff >> 1) << 4) | nl;
        int comp = koff & 1;
        iwpk[((((k4 << 4) + tile) << 5) + lane) * 2 + comp] = val;
    }
}

// ---------------------------------------------------------------------------
// Main LIF step: each 512-thread block (16 waves) handles 16 batch rows.
// Wave w computes the 16x16 output tile for units [16w, 16w+16).
// acc = inputs@W_in + z@W_rec via chained v_wmma_f32_16x16x4_f32.
// ---------------------------------------------------------------------------
__global__ __launch_bounds__(512)
void lif_step(const float* __restrict__ inputs,
              const float* __restrict__ vIn,
              const int*   __restrict__ rIn,
              const float* __restrict__ zIn,
              const v2f*   __restrict__ iwpk,
              const v2f*   __restrict__ wpk,
              float* __restrict__ outV,
              float* __restrict__ outZ,
              int*   __restrict__ outR)
{
    __shared__ float sIn[16][N_IN + 4];    // +4 pad -> conflict-free b64 reads
    __shared__ float sZ [16][UNITS + 4];

    const int m0 = blockIdx.x << 4;

    // Coalesced staging of the 16 activation rows (uniform trip counts).
    for (int idx = threadIdx.x; idx < 16 * N_IN; idx += 512) {
        int row = idx >> 7, col = idx & (N_IN - 1);
        sIn[row][col] = inputs[(size_t)(m0 + row) * N_IN + col];
    }
    for (int idx = threadIdx.x; idx < 16 * UNITS; idx += 512) {
        int row = idx >> 8, col = idx & (UNITS - 1);
        sZ[row][col] = zIn[(size_t)(m0 + row) * UNITS + col];
    }
    __syncthreads();

    const int wave = threadIdx.x >> 5;
    const int lane = threadIdx.x & 31;
    const int half = lane >> 4;        // 0: K0/K1 holder, 1: K2/K3 holder
    const int nl   = lane & 15;        // M (for A) / N (for B,C)
    const int n0   = wave << 4;

    v8f acc = {0.f, 0.f, 0.f, 0.f, 0.f, 0.f, 0.f, 0.f};

    const float* aIn = &sIn[nl][half * 2];
    const float* aZ  = &sZ [nl][half * 2];
    const v2f*   bIn = iwpk + (wave << 5) + lane;
    const v2f*   bW  = wpk  + (wave << 5) + lane;

    // i_in = inputs @ W_in : K = 128 -> 32 WMMA steps
    #pragma unroll 8
    for (int k4 = 0; k4 < N_IN / 4; ++k4) {
        v2f a = *(const v2f*)(aIn + (k4 << 2));     // ds_load_b64 from LDS
        v2f b = bIn[(size_t)k4 << 9];               // global_load_b64, coalesced
        acc = __builtin_amdgcn_wmma_f32_16x16x4_f32(
                  false, a, false, b, (short)0, acc, false, false);
    }

    // i_rec = z @ W_rec : K = 256 -> 64 WMMA steps
    #pragma unroll 8
    for (int k4 = 0; k4 < UNITS / 4; ++k4) {
        v2f a = *(const v2f*)(aZ + (k4 << 2));
        v2f b = bW[(size_t)k4 << 9];
        acc = __builtin_amdgcn_wmma_f32_16x16x4_f32(
                  false, a, false, b, (short)0, acc, false, false);
    }

    // LIF epilogue. C/D layout: VGPR i -> M = i (lanes 0-15) / i+8 (16-31),
    // N = lane & 15.
    #pragma unroll
    for (int i = 0; i < 8; ++i) {
        int rowL = i + (half << 3);           // block-local row
        int row  = m0 + rowL;
        int col  = n0 + nl;
        size_t off = (size_t)row * UNITS + col;

        float itot = acc[i];                  // i_in + i_rec
        float vo   = vIn[off];
        float zo   = sZ[rowL][col];           // reuse staged z (old spikes)
        int   ro   = rIn[off];

        float nv = DECAY_F * vo + itot - zo;  // THR = 1 -> reset = -z
        float vs = nv - 1.0f;                 // (nv - THR)/THR, THR = 1
        float nz = (vs > 0.0f) ? 1.0f : 0.0f;
        if (ro > 0) nz = 0.0f;

        int nr = ro - 1 + ((nz > 0.0f) ? N_REFR : 0);
        nr = nr < 0 ? 0 : (nr > N_REFR ? N_REFR : nr);

        outV[off] = nv;
        outZ[off] = nz;
        outR[off] = nr;
    }
}

// ---------------------------------------------------------------------------
extern "C" void kernel_launch(void* const* d_in, const int* in_sizes, int n_in,
                              void* d_out, int out_size, void* d_ws, size_t ws_size,
                              hipStream_t stream)
{
    const float*         inputs = (const float*)d_in[0];
    const float*         v      = (const float*)d_in[1];
    const int*           r      = (const int*)d_in[2];
    const float*         z      = (const float*)d_in[3];
    const float*         iw     = (const float*)d_in[4];
    const float*         rw     = (const float*)d_in[5];
    const unsigned char* conn   = (const unsigned char*)d_in[6];
    // d_in[7] (ei_mask) and d_in[8] (disconnect_mask) are recomputed on device.

    float* iwpk = (float*)d_ws;                         // 128*256 f32 = 128 KB
    float* wpk  = (float*)d_ws + (size_t)N_IN * UNITS;  // 256*256 f32 = 256 KB

    int prepElems = UNITS * UNITS + N_IN * UNITS;       // 98304
    lif_prep_pack<<<(prepElems + 255) / 256, 256, 0, stream>>>(iw, rw, conn,
                                                               iwpk, wpk);

    int Brows = in_sizes[0] / N_IN;                     // 65536
    size_t plane = (size_t)Brows * UNITS;
    float* outV = (float*)d_out;
    float* outZ = outV + plane;
    int*   outR = (int*)d_out + 2 * plane;

    lif_step<<<Brows / 16, 512, 0, stream>>>(inputs, v, r, z,
                                             (const v2f*)iwpk, (const v2f*)wpk,
                                             outV, outZ, outR);
}